// RGLRUWrapper_31464930410572
// MI455X (gfx1250) — compile-verified
//
#include <hip/hip_runtime.h>
#include <math.h>

typedef float v2f __attribute__((ext_vector_type(2)));
typedef float v8f __attribute__((ext_vector_type(8)));
typedef int   v4i __attribute__((vector_size(4 * sizeof(int))));  // matches builtin param

#define KS 32   // K-slab staged in LDS per iteration
#define TM 64   // block tile rows (m = s*B + b)
#define TN 64   // block tile cols (e)
#define LP 36   // padded LDS row stride: rows 16B-aligned (36*4=144B) and
                // 36 = 4*9 with 9 coprime to 16 -> 16-row column gathers hit
                // 16 distinct banks (conflict-free)

#if __has_builtin(__builtin_amdgcn_global_load_async_to_lds_b128)
#define USE_ASYNC 1
#else
#define USE_ASYNC 0
#endif

#if USE_ASYNC
__device__ __forceinline__ void async_cp16(const float* g, float* l) {
    __builtin_amdgcn_global_load_async_to_lds_b128(
        (__attribute__((address_space(1))) v4i*)(g),
        (__attribute__((address_space(3))) v4i*)(l),
        0, 0);
}
# if __has_builtin(__builtin_amdgcn_s_wait_asynccnt)
#  define WAIT_ASYNC(n) __builtin_amdgcn_s_wait_asynccnt(n)
# else
#  define WAIT_ASYNC(n) asm volatile("s_wait_asynccnt " #n ::: "memory")
# endif
#else
# define WAIT_ASYNC(n)
#endif

__device__ __forceinline__ void stage_load(
    float (*__restrict__ As)[LP], float (*__restrict__ Bxs)[LP],
    float (*__restrict__ Bas)[LP],
    const float* __restrict__ x, const float* __restrict__ Wx,
    const float* __restrict__ Wa,
    int m0, int n0, int k0, int D, int tid)
{
    // 64x32 floats per array = 512 B128 transfers, 2 per thread per array.
    #pragma unroll
    for (int i = 0; i < 2; ++i) {
        const int id  = tid + 256 * i;
        const int row = id >> 3;
        const int c4  = (id & 7) << 2;
#if USE_ASYNC
        async_cp16(x  + (size_t)(m0 + row) * D + k0 + c4, &As [row][c4]);
        async_cp16(Wx + (size_t)(n0 + row) * D + k0 + c4, &Bxs[row][c4]);
        async_cp16(Wa + (size_t)(n0 + row) * D + k0 + c4, &Bas[row][c4]);
#else
        *(float4*)&As [row][c4] = *(const float4*)(x  + (size_t)(m0 + row) * D + k0 + c4);
        *(float4*)&Bxs[row][c4] = *(const float4*)(Wx + (size_t)(n0 + row) * D + k0 + c4);
        *(float4*)&Bas[row][c4] = *(const float4*)(Wa + (size_t)(n0 + row) * D + k0 + c4);
#endif
    }
}

// Fused dual GEMM (fp32 WMMA) + RG-LRU pointwise epilogue.
// Async double-buffered LDS staging: load slab k+1 while WMMAing slab k.
__global__ __launch_bounds__(256) void rg_lru_gates_kernel(
    const float* __restrict__ x,  const float* __restrict__ Wx,
    const float* __restrict__ bx, const float* __restrict__ Wa,
    const float* __restrict__ ba, const float* __restrict__ ap,
    float* __restrict__ y, float* __restrict__ Aout,
    int M, int D, int Bb)
{
    __shared__ float As [2][TM][LP];
    __shared__ float Bxs[2][TN][LP];
    __shared__ float Bas[2][TN][LP];

    const int tid  = threadIdx.x;
    const int lane = tid & 31;
    const int wave = tid >> 5;
    const int hi   = lane >> 4;   // 0: lanes 0-15, 1: lanes 16-31
    const int l16  = lane & 15;

    const int m0 = blockIdx.y * TM;
    const int n0 = blockIdx.x * TN;

    const int r = wave & 3;       // 4 row subtiles of 16
    const int c = wave >> 2;      // 2 col halves of 32

    v8f acc_gx[2] = {};
    v8f acc_ga[2] = {};

    // Prologue: start filling buffer 0.
    stage_load(As[0], Bxs[0], Bas[0], x, Wx, Wa, m0, n0, 0, D, tid);

    int ib = 0;
    for (int k0 = 0; k0 < D; k0 += KS, ib ^= 1) {
        const bool has_next = (k0 + KS) < D;
        if (has_next) {
            // buffer ib^1 was last read in the previous stage; the trailing barrier
            // of that stage makes this overwrite safe.
            stage_load(As[ib ^ 1], Bxs[ib ^ 1], Bas[ib ^ 1],
                       x, Wx, Wa, m0, n0, k0 + KS, D, tid);
            WAIT_ASYNC(6);   // <=6 outstanding -> current stage's 6 landed
        } else {
            WAIT_ASYNC(0);
        }
        __syncthreads();     // current slab visible to all waves

        const float (*__restrict__ A )[LP] = As [ib];
        const float (*__restrict__ BX)[LP] = Bxs[ib];
        const float (*__restrict__ BA)[LP] = Bas[ib];

        #pragma unroll
        for (int kk = 0; kk < KS; kk += 4) {
            // A 16x4 f32 fragment: lane<16 -> K=kk+0..1 ; lane>=16 -> K=kk+2..3
            const int kb = kk + (hi << 1);
            v2f af;
            af.x = A[16 * r + l16][kb];
            af.y = A[16 * r + l16][kb + 1];
            #pragma unroll
            for (int t = 0; t < 2; ++t) {
                // B 4x16 f32 fragment: N = lane%16, same K split as A
                const int nrow = 32 * c + 16 * t + l16;
                v2f bxf, baf;
                bxf.x = BX[nrow][kb];  bxf.y = BX[nrow][kb + 1];
                baf.x = BA[nrow][kb];  baf.y = BA[nrow][kb + 1];
                acc_gx[t] = __builtin_amdgcn_wmma_f32_16x16x4_f32(
                    false, af, false, bxf, (short)0, acc_gx[t], false, false);
                acc_ga[t] = __builtin_amdgcn_wmma_f32_16x16x4_f32(
                    false, af, false, baf, (short)0, acc_ga[t], false, false);
            }
        }
        __syncthreads();     // all waves done reading before next overwrite
    }

    // Epilogue: C/D layout = VGPR j -> M = j (lanes 0-15) / j+8 (lanes 16-31),
    // N = lane%16. Intermediates are streamed (read once by the scan kernel):
    // store non-temporally so they don't evict the L2-resident weights.
    #pragma unroll
    for (int t = 0; t < 2; ++t) {
        const int n     = n0 + 32 * c + 16 * t + l16;
        const float bxn = bx[n];
        const float ban = ba[n];
        const float sp  = log1pf(__expf(ap[n]));   // softplus(a_param[e])
        #pragma unroll
        for (int j = 0; j < 8; ++j) {
            const int m = m0 + 16 * r + j + 8 * hi;
            const float gx = 1.0f / (1.0f + __expf(-(acc_gx[t][j] + bxn)));
            const float ga = 1.0f / (1.0f + __expf(-(acc_ga[t][j] + ban)));
            const float la = -8.0f * ga * sp;           // log_a
            float a    = __expf(la);
            float mult = sqrtf(fmaxf(0.0f, 1.0f - __expf(2.0f * la)));
            if (m < Bb) { a = 0.0f; mult = 1.0f; }      // reset at s == 0
            const size_t idx = (size_t)m * D + n;
            __builtin_nontemporal_store(x[idx] * gx * mult, y + idx); // normed_x
            __builtin_nontemporal_store(a, Aout + idx);
        }
    }
}

// Diagonal linear recurrence, in place over y: h = a*h + normed_x.
// Only ~512 waves exist (one channel per lane) -> latency hiding must come from
// ILP: deep unroll lets the compiler hoist ~32 independent loads per wave.
// All traffic is single-use streaming -> non-temporal.
__global__ __launch_bounds__(256) void rg_lru_scan_kernel(
    const float* __restrict__ Aarr, const float* __restrict__ h0,
    float* __restrict__ y, float* __restrict__ hn, int S, int BD)
{
    const int cidx = blockIdx.x * 256 + threadIdx.x;
    if (cidx >= BD) return;
    float h = h0[cidx];
    size_t idx = (size_t)cidx;
    #pragma unroll 16
    for (int s = 0; s < S; ++s) {
        const float a = __builtin_nontemporal_load(Aarr + idx);
        const float v = __builtin_nontemporal_load(y + idx);
        h = fmaf(a, h, v);
        __builtin_nontemporal_store(h, y + idx);
        idx += (size_t)BD;
    }
    hn[cidx] = h;
}

extern "C" void kernel_launch(void* const* d_in, const int* in_sizes, int n_in,
                              void* d_out, int out_size, void* d_ws, size_t ws_size,
                              hipStream_t stream) {
    const float* x  = (const float*)d_in[0];
    const float* h0 = (const float*)d_in[1];
    const float* Wx = (const float*)d_in[2];
    const float* bx = (const float*)d_in[3];
    const float* Wa = (const float*)d_in[4];
    const float* ba = (const float*)d_in[5];
    const float* ap = (const float*)d_in[6];

    const int D  = in_sizes[6];                 // width
    const int Bb = in_sizes[1] / D;             // batch
    const int S  = in_sizes[0] / in_sizes[1];   // seq len
    const int M  = S * Bb;                      // GEMM rows

    float* y    = (float*)d_out;                // (S,B,D) output block
    float* hn   = y + (size_t)M * D;            // final state block
    float* Aarr = (float*)d_ws;                 // decay coefficients scratch

    dim3 grid(D / TN, M / TM);
    rg_lru_gates_kernel<<<grid, 256, 0, stream>>>(x, Wx, bx, Wa, ba, ap,
                                                  y, Aarr, M, D, Bb);

    const int BD = Bb * D;
    rg_lru_scan_kernel<<<(BD + 255) / 256, 256, 0, stream>>>(Aarr, h0, y, hn, S, BD);
}